// VariantCoeLinear1d_82291573391976
// MI455X (gfx1250) — compile-verified
//
#include <hip/hip_runtime.h>
#include <stdint.h>

#define BATCH 128
#define N 4000
#define NA 400
#define NTHREADS 512

#ifndef __has_builtin
#define __has_builtin(x) 0
#endif

#if __has_builtin(__builtin_amdgcn_global_load_async_to_lds_b128)
#define HAVE_ASYNC_LOAD 1
#else
#define HAVE_ASYNC_LOAD 0
#endif

#if __has_builtin(__builtin_amdgcn_s_wait_asynccnt)
#define WAIT_ASYNC() __builtin_amdgcn_s_wait_asynccnt(0)
#else
#define WAIT_ASYNC() asm volatile("s_wait_asynccnt 0" ::: "memory")
#endif

// GCC-style vector type matching the async builtin's expected pointee
typedef int v4i __attribute__((vector_size(4 * sizeof(int))));
typedef __attribute__((address_space(1))) v4i* gv4i_p;   // global (AS1) v4i*
typedef __attribute__((address_space(3))) v4i* lv4i_p;   // LDS    (AS3) v4i*

// f(u) = 0.5*u*(3-u^2) + 25*u^2*(0.75 - 2u + 1.5u^2 - 0.25u^4)
__device__ __forceinline__ float f_eval(float u) {
    float u2 = u * u;
    float p  = 0.75f - 2.0f * u + 1.5f * u2 - 0.25f * u2 * u2;
    return 0.5f * u * (3.0f - u2) + 25.0f * u2 * p;
}

// f'(u) = 1.5 - 1.5u^2 + 25*(1.5u - 6u^2 + 6u^3 - 1.5u^5)
__device__ __forceinline__ float fp_eval(float u) {
    float u2 = u * u;
    float t  = 1.5f * u - 6.0f * u2 + 6.0f * u2 * u - 1.5f * u2 * u2 * u;
    return 1.5f - 1.5f * u2 + 25.0f * t;
}

// jnp.interp: clamp outside [ua0, uaN]; linear inside.
// Fast path: direct index guess (u_a is uniform in practice) + monotone walk
// fixup, which is exact for any monotone u_a. Cuts the dependent-LDS chain
// from ~9 (binary search) to ~2.
__device__ __forceinline__ float interpA(float x, const float* sua, const float* sAa,
                                         float ua0, float uaN, float idx_scale) {
    if (x <= ua0) return sAa[0];
    if (x >= uaN) return sAa[NA];
    int g = (int)((x - ua0) * idx_scale);
    if (g < 0) g = 0;
    if (g > NA - 1) g = NA - 1;
    // restore invariant sua[g] <= x < sua[g+1]  (0-2 iterations in practice)
    while (g > 0 && x < sua[g]) --g;
    while (g < NA - 1 && x >= sua[g + 1]) ++g;
    float xl = sua[g], xr = sua[g + 1];
    float yl = sAa[g], yr = sAa[g + 1];
    float t = (x - xl) / (xr - xl);
    return yl + t * (yr - yl);
}

__global__ __launch_bounds__(NTHREADS)
void coe_linear1d_kernel(const float* __restrict__ init,
                         const float* __restrict__ u_a,
                         const float* __restrict__ dt_p,
                         const float* __restrict__ dx_p,
                         const int*   __restrict__ step_p,
                         float*       __restrict__ out) {
    __shared__ float ubuf[2][N];      // double-buffered state row (LDS-resident)
    __shared__ float AU[N];           // A(u) per grid point
    __shared__ float sua[NA + 1];     // u_a row
    __shared__ float sAa[NA + 1];     // cumtrapz table

    const int b   = blockIdx.x;
    const int tid = threadIdx.x;
    const float dt = dt_p[0];
    const float dx = dx_p[0];
    const int steps = step_p[0];
    const float r = dt / dx;          // dt/dx
    const float q = dt / (dx * dx);   // dt/dx^2

    const float* grow = init + (size_t)b * N;

    // ---- initial row: gfx1250 async global->LDS DMA path ----
#if HAVE_ASYNC_LOAD
    for (int v = tid; v < N / 4; v += NTHREADS) {
        __builtin_amdgcn_global_load_async_to_lds_b128(
            (gv4i_p)(grow + 4 * v),
            (lv4i_p)(&ubuf[0][4 * v]),
            0, 0);
    }
    WAIT_ASYNC();
#else
    for (int i = tid; i < N; i += NTHREADS) ubuf[0][i] = grow[i];
#endif

    // ---- u_a table into LDS ----
    const float* garow = u_a + (size_t)b * (NA + 1);
    for (int k = tid; k <= NA; k += NTHREADS) sua[k] = garow[k];

    __syncthreads();

    // ---- A_a = cumtrapz(0.5*a_real(u_a), u_a), serial (once per block) ----
    if (tid == 0) {
        float acc = 0.0f;
        sAa[0] = 0.0f;
        float pu = sua[0];
        float pa = 0.5f * (pu < 0.5f ? 0.0f : 1.0f);
        for (int j = 1; j <= NA; ++j) {
            float cu = sua[j];
            float ca = 0.5f * (cu < 0.5f ? 0.0f : 1.0f);
            acc += 0.5f * (ca + pa) * (cu - pu);
            sAa[j] = acc;
            pu = cu; pa = ca;
        }
    }

    // ---- traj[0] = init row ----
    {
        float* orow0 = out + (size_t)b * N;
        for (int i = tid; i < N; i += NTHREADS) orow0[i] = ubuf[0][i];
    }
    __syncthreads();

    // register-cached interp parameters (uniform across the block)
    const float ua0 = sua[0];
    const float uaN = sua[NA];
    const float idx_scale = (float)NA / (uaN - ua0);

    // ---- time loop, fully LDS-resident ----
    int p = 0;
    for (int s = 0; s < steps; ++s) {
        const float* cur = ubuf[p];
        float*       nxt = ubuf[1 - p];

        // phase 1: pointwise A(u) table lookup
        for (int i = tid; i < N; i += NTHREADS)
            AU[i] = interpA(cur[i], sua, sAa, ua0, uaN, idx_scale);
        __syncthreads();

        // phase 2: flux + stencil update, write LDS + trajectory stream
        float* orow = out + (size_t)(s + 1) * (BATCH * (size_t)N) + (size_t)b * N;
        for (int i = tid; i < N; i += NTHREADS) {
            if (i >= 1 && i <= N - 2) {
                float uim = cur[i - 1], ui = cur[i], uip = cur[i + 1];
                float Fim = f_eval(uim), Fi = f_eval(ui), Fip = f_eval(uip);
                float Gim = fabsf(fp_eval(uim));
                float Gi  = fabsf(fp_eval(ui));
                float Gip = fabsf(fp_eval(uip));
                float fhR = 0.5f * (Fi + Fip) - 0.5f * fmaxf(Gi, Gip) * (uip - ui);
                float fhL = 0.5f * (Fim + Fi) - 0.5f * fmaxf(Gim, Gi) * (ui - uim);
                float lap = AU[i + 1] - 2.0f * AU[i] + AU[i - 1];
                float un  = ui - r * (fhR - fhL) + q * lap;
                nxt[i]  = un;
                orow[i] = un;
                if (i == 1)     { nxt[0] = un;     orow[0] = un; }
                if (i == N - 2) { nxt[N - 1] = un; orow[N - 1] = un; }
            }
        }
        __syncthreads();
        p ^= 1;
    }
}

extern "C" void kernel_launch(void* const* d_in, const int* in_sizes, int n_in,
                              void* d_out, int out_size, void* d_ws, size_t ws_size,
                              hipStream_t stream) {
    const float* init   = (const float*)d_in[0];
    const float* u_a    = (const float*)d_in[1];
    const float* dt     = (const float*)d_in[2];
    const float* dx     = (const float*)d_in[3];
    const int*   stepnp = (const int*)d_in[4];
    float* out = (float*)d_out;

    hipLaunchKernelGGL(coe_linear1d_kernel, dim3(BATCH), dim3(NTHREADS), 0, stream,
                       init, u_a, dt, dx, stepnp, out);
}